// SimpleGraphSAGE_10514079941027
// MI455X (gfx1250) — compile-verified
//
#include <hip/hip_runtime.h>

typedef float v2f __attribute__((ext_vector_type(2)));
typedef float v8f __attribute__((ext_vector_type(8)));

#define NN 50000
#define NE 1600000
#define DD 128

// ---------------- zero scratch ----------------
__global__ void sage_zero_kernel(float* __restrict__ p, long n) {
  long i = (long)blockIdx.x * blockDim.x + threadIdx.x;
  long stride = (long)gridDim.x * blockDim.x;
  for (; i < n; i += stride) p[i] = 0.0f;
}

// ---------------- edge scatter: sum[dst] += feat[src]; cnt[dst] += 1 -------
// one wave (32 lanes) per edge, 4 floats per lane (DD = 128 = 32*4)
__global__ __launch_bounds__(256)
void sage_scatter_kernel(const float* __restrict__ feat,
                         const int* __restrict__ src,
                         const int* __restrict__ dst,
                         float* __restrict__ sum,
                         float* __restrict__ cnt,   // may be nullptr
                         int n_edges) {
  int edge = (int)((blockIdx.x * blockDim.x + threadIdx.x) >> 5);
  int lane = threadIdx.x & 31;
  if (edge >= n_edges) return;
  int s = src[edge];
  int d = dst[edge];
  float4 v = ((const float4*)(feat + (size_t)s * DD))[lane];
  float* op = sum + (size_t)d * DD + lane * 4;
  unsafeAtomicAdd(op + 0, v.x);
  unsafeAtomicAdd(op + 1, v.y);
  unsafeAtomicAdd(op + 2, v.z);
  unsafeAtomicAdd(op + 3, v.w);
  if (cnt != nullptr && lane == 0) unsafeAtomicAdd(cnt + d, 1.0f);
}

// ---------------- fused SAGE layer GEMM -----------------------------------
// out[16-row strip] = (neigh_sum/max(cnt,1)) @ Wl^T + bias + self @ Wr^T
// One wave per 16-row strip; 8 column tiles of 16 -> 8 x v8f accumulators.
// V_WMMA_F32_16X16X4_F32:
//   A 16x4: lanes 0-15 K=0..1 in v[0..1], lanes 16-31 K=2..3
//   B 4x16: same K striping, N = laneLo
//   C/D 16x16: lanes 0-15 -> M=0..7 in v[0..7], lanes 16-31 -> M=8..15
__global__ __launch_bounds__(256)
void sage_gemm_kernel(const float* __restrict__ neigh_sum,
                      const float* __restrict__ cnt,
                      const float* __restrict__ self_feat,
                      const float* __restrict__ Wl,
                      const float* __restrict__ Wr,
                      const float* __restrict__ bias,
                      float* __restrict__ out,
                      int relu) {
  int wave   = (int)(blockIdx.x * (blockDim.x >> 5) + (threadIdx.x >> 5));
  int lane   = threadIdx.x & 31;
  int laneLo = lane & 15;
  int hi     = lane >> 4;          // 0: K=0..1 / M=0..7, 1: K=2..3 / M=8..15
  int rowbase = wave * 16;
  if (rowbase >= NN) return;       // wave-uniform: EXEC stays all-ones
  int row = rowbase + laneLo;

  v8f acc[8] = {};

  float scale = 1.0f / fmaxf(cnt[row], 1.0f);

  const float* aN = neigh_sum + (size_t)row * DD + 2 * hi;
  const float* aS = self_feat + (size_t)row * DD + 2 * hi;

  // pass 1: neighbor mean x Wl^T   (B[k][j] = Wl[j][k], row-major Wl[j*DD+k])
  for (int k = 0; k < DD; k += 4) {
    v2f a = *(const v2f*)(aN + k);
    a *= scale;
#pragma unroll
    for (int c = 0; c < 8; ++c) {
      v2f b = *(const v2f*)(Wl + (size_t)(c * 16 + laneLo) * DD + k + 2 * hi);
      acc[c] = __builtin_amdgcn_wmma_f32_16x16x4_f32(
          false, a, false, b, (short)0, acc[c], false, false);
    }
  }
  // pass 2: self features x Wr^T
  for (int k = 0; k < DD; k += 4) {
    v2f a = *(const v2f*)(aS + k);
#pragma unroll
    for (int c = 0; c < 8; ++c) {
      v2f b = *(const v2f*)(Wr + (size_t)(c * 16 + laneLo) * DD + k + 2 * hi);
      acc[c] = __builtin_amdgcn_wmma_f32_16x16x4_f32(
          false, a, false, b, (short)0, acc[c], false, false);
    }
  }

  // bias + optional relu + store (D-matrix layout above)
#pragma unroll
  for (int c = 0; c < 8; ++c) {
    float bv = bias[c * 16 + laneLo];
#pragma unroll
    for (int i = 0; i < 8; ++i) {
      float v = acc[c][i] + bv;
      if (relu) v = fmaxf(v, 0.0f);
      out[(size_t)(rowbase + hi * 8 + i) * DD + c * 16 + laneLo] = v;
    }
  }
}

extern "C" void kernel_launch(void* const* d_in, const int* in_sizes, int n_in,
                              void* d_out, int out_size, void* d_ws, size_t ws_size,
                              hipStream_t stream) {
  const float* x   = (const float*)d_in[0];
  const int*   ei  = (const int*)d_in[1];   // (2, NE): row 0 = src, row 1 = dst
  const float* W1l = (const float*)d_in[2];
  const float* b1  = (const float*)d_in[3];
  const float* W1r = (const float*)d_in[4];
  const float* W2l = (const float*)d_in[5];
  const float* b2  = (const float*)d_in[6];
  const float* W2r = (const float*)d_in[7];
  const int* src = ei;
  const int* dst = ei + NE;
  float* out = (float*)d_out;

  float* sum = (float*)d_ws;                  // NN*DD
  float* cnt = sum + (size_t)NN * DD;         // NN (contiguous after sum)
  float* h   = cnt + NN;                      // NN*DD

  const long sum_cnt_elems = (long)NN * DD + NN;
  const int  zero_blocks   = 1024;
  const int  strips        = NN / 16;                    // 3125 (exact)
  const int  gemm_blocks   = (strips + 7) / 8;           // 8 waves/block
  const int  scat_blocks   = (NE * 32 + 255) / 256;      // 1 wave per edge

  // ---- layer 1 ----
  sage_zero_kernel<<<zero_blocks, 256, 0, stream>>>(sum, sum_cnt_elems);
  sage_scatter_kernel<<<scat_blocks, 256, 0, stream>>>(x, src, dst, sum, cnt, NE);
  sage_gemm_kernel<<<gemm_blocks, 256, 0, stream>>>(sum, cnt, x, W1l, W1r, b1, h, 1);

  // ---- layer 2 (reuse cnt) ----
  sage_zero_kernel<<<zero_blocks, 256, 0, stream>>>(sum, (long)NN * DD);
  sage_scatter_kernel<<<scat_blocks, 256, 0, stream>>>(h, src, dst, sum, nullptr, NE);
  sage_gemm_kernel<<<gemm_blocks, 256, 0, stream>>>(sum, cnt, h, W2l, W2r, b2, out, 0);
}